// HybridModel_53979148976318
// MI455X (gfx1250) — compile-verified
//
#include <hip/hip_runtime.h>

typedef __attribute__((ext_vector_type(2))) float v2f;
typedef __attribute__((ext_vector_type(8))) float v8f;

#define BETA_C  0.95f
#define THR_C   1.0f
#define NSTEPS  100
#define SHID    1024
#define NHID    1024
#define CO      256
#define NOUT    64
#define COMBK   384
#define TB      16
#define XROW    105

__device__ __forceinline__ v8f wmma4(v2f a, v2f b, v8f c) {
    // V_WMMA_F32_16X16X4_F32 : D(16x16,f32) = A(16x4,f32) * B(4x16,f32) + C
    return __builtin_amdgcn_wmma_f32_16x16x4_f32(
        false, a, false, b, (short)0, c, false, false);
}

__device__ __forceinline__ v8f splat8(float v) {
    v8f r = {v, v, v, v, v, v, v, v};
    return r;
}

// ---------------------------------------------------------------------------
// out[b,o] = comb_b[o]   (branch kernels accumulate on top)
// ---------------------------------------------------------------------------
__global__ void init_out_kernel(float* __restrict__ out,
                                const float* __restrict__ comb_b, int total) {
    int i = blockIdx.x * blockDim.x + threadIdx.x;
    if (i < total) out[i] = comb_b[i & (NOUT - 1)];
}

// ---------------------------------------------------------------------------
// Fused combine step: out_tile(16x64) += Atile(16xK,LDS) @ comb_w[:,off:off+K]^T
// One wave per 16-col N-tile (waves 0..3). A row-major in LDS with stride K.
// ---------------------------------------------------------------------------
__device__ __forceinline__ void combine_add(const float* __restrict__ s_a, int kdim,
                                            const float* __restrict__ comb_w, int comb_off,
                                            float* __restrict__ out, int row0,
                                            int half, int idx, int nbase) {
    v8f d;
#pragma unroll
    for (int r = 0; r < 8; ++r)
        d[r] = out[(size_t)(row0 + r + 8 * half) * NOUT + nbase + idx];

    const float* brow = comb_w + (size_t)(nbase + idx) * COMBK + comb_off;
    const float* arow = s_a + idx * kdim;
#pragma unroll 4
    for (int kk = 0; kk < kdim / 4; ++kk) {
        int k0 = kk * 4 + half * 2;
        v2f a = *reinterpret_cast<const v2f*>(arow + k0);
        v2f b = *reinterpret_cast<const v2f*>(brow + k0);
        d = wmma4(a, b, d);
    }
#pragma unroll
    for (int r = 0; r < 8; ++r)
        out[(size_t)(row0 + r + 8 * half) * NOUT + nbase + idx] = d[r];
}

// ---------------------------------------------------------------------------
// SNN branch: LIF recurrence, 100 steps. mem1/spk1 resident in LDS; layer-2
// state resident in WMMA accumulator registers. 4 waves / 16 rows per WG.
// Spikes are binarized once per step in the update phase, so the WMMA loop's
// A-operand is a plain ds_load_b64 (no VALU on the critical path).
// ---------------------------------------------------------------------------
__global__ __launch_bounds__(128)
void snn_kernel(const float* __restrict__ x,
                const float* __restrict__ w1, const float* __restrict__ b1,
                const float* __restrict__ w2, const float* __restrict__ b2,
                const float* __restrict__ comb_w,
                float* __restrict__ out) {
    __shared__ float s_cur1[TB * SHID];   // 64 KB
    __shared__ float s_mem1[TB * SHID];   // 64 KB
    __shared__ float s_spk1[TB * SHID];   // 64 KB
    __shared__ float s_feat[TB * 5];
    __shared__ float s_res[TB * NOUT];    // 4 KB

    const int tid  = threadIdx.x;
    const int lane = tid & 31;
    const int wave = tid >> 5;      // 0..3 -> N-tile
    const int half = lane >> 4;
    const int idx  = lane & 15;
    const int row0 = blockIdx.x * TB;

    for (int i = tid; i < TB * 5; i += 128) {
        int b = i / 5, c = i % 5;
        s_feat[i] = x[(size_t)(row0 + b) * XROW + c];
    }
    __syncthreads();

    // cur1 = feats @ w1^T + b1 ; mem1 = 0
    for (int i = tid; i < TB * SHID; i += 128) {
        int b = i >> 10, h = i & (SHID - 1);
        const float* wr = w1 + h * 5;
        const float* f  = s_feat + b * 5;
        float s = b1[h] + f[0]*wr[0] + f[1]*wr[1] + f[2]*wr[2]
                        + f[3]*wr[3] + f[4]*wr[4];
        s_cur1[i] = s;
        s_mem1[i] = 0.0f;
    }

    const int nbase = wave * 16;
    const float bias2 = b2[nbase + idx];
    const float* w2row = w2 + (size_t)(nbase + idx) * SHID;
    __builtin_prefetch(w2row, 0, 0);   // global_prefetch_b8: warm this wave's W2 slice

    v8f mem2 = splat8(0.0f);
    v8f spk2 = splat8(0.0f);
    v8f acc  = splat8(0.0f);

    const float* arow = s_spk1 + idx * SHID;

    for (int t = 0; t < NSTEPS; ++t) {
        __syncthreads();
        // mem1 <- beta*mem1 + cur1 - spk_prev*THR ; spk1 <- (mem1_new > THR)
        // (spk_prev re-derived from stored mem1; pairwise for b64 LDS traffic)
        for (int i = tid * 2; i < TB * SHID; i += 256) {
            v2f m  = *reinterpret_cast<v2f*>(s_mem1 + i);
            v2f cu = *reinterpret_cast<const v2f*>(s_cur1 + i);
            v2f mn, sn;
            mn.x = BETA_C * m.x + cu.x - ((m.x > THR_C) ? THR_C : 0.0f);
            mn.y = BETA_C * m.y + cu.y - ((m.y > THR_C) ? THR_C : 0.0f);
            sn.x = (mn.x > THR_C) ? 1.0f : 0.0f;
            sn.y = (mn.y > THR_C) ? 1.0f : 0.0f;
            *reinterpret_cast<v2f*>(s_mem1 + i) = mn;
            *reinterpret_cast<v2f*>(s_spk1 + i) = sn;
        }
        __syncthreads();

        // cur2 tile = spk1(16x1024) @ w2^T(1024x16) + b2
        v8f c = splat8(bias2);
#pragma unroll 4
        for (int kk = 0; kk < SHID / 4; ++kk) {
            int h0 = kk * 4 + half * 2;
            v2f a = *reinterpret_cast<const v2f*>(arow + h0);
            v2f b = *reinterpret_cast<const v2f*>(w2row + h0);
            c = wmma4(a, b, c);
        }
        // layer-2 LIF entirely in accumulator registers
#pragma unroll
        for (int r = 0; r < 8; ++r) {
            float m = BETA_C * mem2[r] + c[r] - spk2[r] * THR_C;
            float s = (m > THR_C) ? 1.0f : 0.0f;
            mem2[r] = m; spk2[r] = s; acc[r] += s;
        }
    }

    // snn_out tile -> LDS (spk mean over steps)
#pragma unroll
    for (int r = 0; r < 8; ++r)
        s_res[(r + 8 * half) * NOUT + nbase + idx] = acc[r] / (float)NSTEPS;
    __syncthreads();

    combine_add(s_res, NOUT, comb_w, /*off=*/0, out, row0, half, idx, nbase);
}

// ---------------------------------------------------------------------------
// NN branch: relu(feats@w1^T+b1) @ w2^T + b2, fused combine.
// ---------------------------------------------------------------------------
__global__ __launch_bounds__(128)
void nn_kernel(const float* __restrict__ x,
               const float* __restrict__ w1, const float* __restrict__ b1,
               const float* __restrict__ w2, const float* __restrict__ b2,
               const float* __restrict__ comb_w,
               float* __restrict__ out) {
    __shared__ float s_hid[TB * NHID];    // 64 KB
    __shared__ float s_feat[TB * 5];
    __shared__ float s_res[TB * NOUT];

    const int tid  = threadIdx.x;
    const int lane = tid & 31;
    const int wave = tid >> 5;
    const int half = lane >> 4;
    const int idx  = lane & 15;
    const int row0 = blockIdx.x * TB;

    for (int i = tid; i < TB * 5; i += 128) {
        int b = i / 5, c = i % 5;
        s_feat[i] = x[(size_t)(row0 + b) * XROW + c];
    }
    __syncthreads();

    for (int i = tid; i < TB * NHID; i += 128) {
        int b = i >> 10, h = i & (NHID - 1);
        const float* wr = w1 + h * 5;
        const float* f  = s_feat + b * 5;
        float s = b1[h] + f[0]*wr[0] + f[1]*wr[1] + f[2]*wr[2]
                        + f[3]*wr[3] + f[4]*wr[4];
        s_hid[i] = fmaxf(s, 0.0f);
    }
    __syncthreads();

    const int nbase = wave * 16;
    const float* w2row = w2 + (size_t)(nbase + idx) * NHID;
    const float* arow  = s_hid + idx * NHID;

    v8f c = splat8(b2[nbase + idx]);
#pragma unroll 4
    for (int kk = 0; kk < NHID / 4; ++kk) {
        int h0 = kk * 4 + half * 2;
        v2f a = *reinterpret_cast<const v2f*>(arow + h0);
        v2f b = *reinterpret_cast<const v2f*>(w2row + h0);
        c = wmma4(a, b, c);
    }
#pragma unroll
    for (int r = 0; r < 8; ++r)
        s_res[(r + 8 * half) * NOUT + nbase + idx] = c[r];
    __syncthreads();

    combine_add(s_res, NOUT, comb_w, /*off=*/NOUT, out, row0, half, idx, nbase);
}

// ---------------------------------------------------------------------------
// CNN branch: conv1d(k=3,pad=1) -> relu -> maxpool2 -> FC(1600->256), fused
// combine. 8 waves: 16 N-tiles of the FC split 2-per-wave.
// ---------------------------------------------------------------------------
__global__ __launch_bounds__(256)
void cnn_kernel(const float* __restrict__ x,
                const float* __restrict__ conv_w, const float* __restrict__ conv_b,
                const float* __restrict__ fc_w, const float* __restrict__ fc_b,
                const float* __restrict__ comb_w,
                float* __restrict__ out) {
    __shared__ float s_wave[TB * 100];    // 6.4 KB
    __shared__ float s_pool[TB * 1600];   // 102.4 KB
    __shared__ float s_res[TB * CO];      // 16 KB

    const int tid  = threadIdx.x;
    const int lane = tid & 31;
    const int wave = tid >> 5;      // 0..7
    const int half = lane >> 4;
    const int idx  = lane & 15;
    const int row0 = blockIdx.x * TB;

    for (int i = tid; i < TB * 100; i += 256) {
        int b = i / 100, p = i % 100;
        s_wave[i] = x[(size_t)(row0 + b) * XROW + 5 + p];
    }
    __syncthreads();

    // conv + relu + maxpool-2 -> pooled[b][ch*50+p2]
    for (int i = tid; i < TB * 1600; i += 256) {
        int b  = i / 1600;
        int r  = i % 1600;
        int ch = r / 50, p2 = r % 50;
        const float* wv = s_wave + b * 100;
        float cw0 = conv_w[ch * 3 + 0];
        float cw1 = conv_w[ch * 3 + 1];
        float cw2 = conv_w[ch * 3 + 2];
        float cb  = conv_b[ch];
        float m = 0.0f;   // relu outputs are >= 0
#pragma unroll
        for (int q = 0; q < 2; ++q) {
            int p = 2 * p2 + q;
            float s = cb + cw1 * wv[p];
            if (p >= 1)  s += cw0 * wv[p - 1];
            if (p <= 98) s += cw2 * wv[p + 1];
            m = fmaxf(m, fmaxf(s, 0.0f));
        }
        s_pool[b * 1600 + ch * 50 + p2] = m;
    }
    __syncthreads();

    // FC: pooled(16x1600) @ fc_w^T(1600x256) + fc_b
    const float* arow = s_pool + idx * 1600;
    for (int nt = wave; nt < 16; nt += 8) {
        int nbase = nt * 16;
        const float* brow = fc_w + (size_t)(nbase + idx) * 1600;
        v8f c = splat8(fc_b[nbase + idx]);
#pragma unroll 4
        for (int kk = 0; kk < 1600 / 4; ++kk) {
            int h0 = kk * 4 + half * 2;
            v2f a = *reinterpret_cast<const v2f*>(arow + h0);
            v2f b = *reinterpret_cast<const v2f*>(brow + h0);
            c = wmma4(a, b, c);
        }
#pragma unroll
        for (int r = 0; r < 8; ++r)
            s_res[(r + 8 * half) * CO + nbase + idx] = c[r];
    }
    __syncthreads();

    if (wave < 4)
        combine_add(s_res, CO, comb_w, /*off=*/2 * NOUT, out, row0, half, idx, wave * 16);
}

// ---------------------------------------------------------------------------
extern "C" void kernel_launch(void* const* d_in, const int* in_sizes, int n_in,
                              void* d_out, int out_size, void* d_ws, size_t ws_size,
                              hipStream_t stream) {
    const float* x        = (const float*)d_in[0];
    const float* snn_w1   = (const float*)d_in[1];
    const float* snn_b1   = (const float*)d_in[2];
    const float* snn_w2   = (const float*)d_in[3];
    const float* snn_b2   = (const float*)d_in[4];
    const float* nn_w1    = (const float*)d_in[5];
    const float* nn_b1    = (const float*)d_in[6];
    const float* nn_w2    = (const float*)d_in[7];
    const float* nn_b2    = (const float*)d_in[8];
    const float* conv_w   = (const float*)d_in[9];
    const float* conv_b   = (const float*)d_in[10];
    const float* cnn_fc_w = (const float*)d_in[11];
    const float* cnn_fc_b = (const float*)d_in[12];
    const float* comb_w   = (const float*)d_in[13];
    const float* comb_b   = (const float*)d_in[14];
    float* out = (float*)d_out;

    const int B = in_sizes[0] / XROW;     // 16384
    const int total = B * NOUT;
    const int ntiles = B / TB;            // 1024 workgroups

    init_out_kernel<<<(total + 255) / 256, 256, 0, stream>>>(out, comb_b, total);
    snn_kernel<<<ntiles, 128, 0, stream>>>(x, snn_w1, snn_b1, snn_w2, snn_b2, comb_w, out);
    nn_kernel <<<ntiles, 128, 0, stream>>>(x, nn_w1, nn_b1, nn_w2, nn_b2, comb_w, out);
    cnn_kernel<<<ntiles, 256, 0, stream>>>(x, conv_w, conv_b, cnn_fc_w, cnn_fc_b, comb_w, out);
}